// PQC_6794638263043
// MI455X (gfx1250) — compile-verified
//
#include <hip/hip_runtime.h>
#include <math.h>

// ---------------------------------------------------------------------------
// Out = U @ X @ U^H,  U = kron(uq0..uq11), DIM = 4096.
// Factor U = U_hi (qubits 0-5, 64x64) ⊗ U_lo (qubits 6-11, 64x64).
// Since X is real:  Out = U * (U * X^T)^H      (two identical left-multiplies)
//   pass A: W  = U * X^T      (lo-stage reads X transposed -> contiguous blocks)
//   pass B: Out= U * W^H      (lo-stage reads W conj-transposed -> contiguous)
// Each left-multiply = lo-stage (rows h*64+l', l' contiguous) then
// hi-stage (rows h'*64+l, stride-64 gather, 128B contiguous per row).
// Inner math: per-wave 16x16 complex tile, K=64, via V_WMMA_F32_16X16X4_F32.
// Traffic ~1.0 GB over 4 passes -> ~45us at 23.3 TB/s; compute ~30 GFLOP f32
// on WMMA is negligible. Intermediate (134MB) is L2-resident (192MB L2).
// ---------------------------------------------------------------------------

#define DIM 4096
#define W_MUL 0.63245553203367586640f /* sqrt(2)/sqrt(5) */

typedef float v2f __attribute__((ext_vector_type(2)));
typedef float v8f __attribute__((ext_vector_type(8)));

// D = A(16x4 f32) x B(4x16 f32) + C(16x16 f32), wave32 WMMA.
// 8-arg form: (neg_a, A, neg_b, B, c_mod, C, reuse_a, reuse_b)
__device__ __forceinline__ v8f wmma4(v2f a, v2f b, v8f c) {
  return __builtin_amdgcn_wmma_f32_16x16x4_f32(false, a, false, b, (short)0, c,
                                               false, false);
}

// ---------------------------------------------------------------------------
// Build U_lo / U_hi (64x64 complex each) analytically from the 36 angles.
// gates layout (floats): [0)Glo_r[4096] [4096)Glo_i [8192)Ghi_r [12288)Ghi_i
// Kron entry [r,c] = prod_t uq[qbase+t][ (r>>(5-t))&1, (c>>(5-t))&1 ].
// uq = RZ(w2) @ RY(w1) @ RX(w0), half-angles, weights pre-scaled by W_MUL.
// ---------------------------------------------------------------------------
__global__ void build_gates_kernel(const float* __restrict__ w,
                                   float* __restrict__ g) {
  int tid = blockIdx.x * blockDim.x + threadIdx.x; // 0..8191
  if (tid >= 8192) return;
  int mat = tid >> 12;     // 0 = U_lo (qubits 6..11), 1 = U_hi (qubits 0..5)
  int e = tid & 4095;
  int r = e >> 6;
  int c = e & 63;
  int qbase = mat ? 0 : 6;
  float pr = 1.0f, pi = 0.0f;
#pragma unroll
  for (int t = 0; t < 6; ++t) {
    int q = qbase + t;
    int rb = (r >> (5 - t)) & 1;
    int cb = (c >> (5 - t)) & 1;
    float hx = 0.5f * W_MUL * w[3 * q + 0];
    float hy = 0.5f * W_MUL * w[3 * q + 1];
    float hz = 0.5f * W_MUL * w[3 * q + 2];
    float cx, sx, cy, sy, cz, sz;
    sincosf(hx, &sx, &cx);
    sincosf(hy, &sy, &cy);
    sincosf(hz, &sz, &cz);
    // m = RY @ RX entry [rb][cb]
    float mr, mi;
    if (rb == 0) {
      if (cb == 0) { mr =  cy * cx; mi =  sy * sx; }
      else         { mr = -sy * cx; mi = -cy * sx; }
    } else {
      if (cb == 0) { mr =  sy * cx; mi = -cy * sx; }
      else         { mr =  cy * cx; mi = -sy * sx; }
    }
    // RZ row factor: row0 -> e^{-i hz}, row1 -> e^{+i hz}
    float zr = cz;
    float zi = (rb == 0) ? -sz : sz;
    float er = zr * mr - zi * mi;
    float ei = zr * mi + zi * mr;
    float npr = pr * er - pi * ei;
    float npi = pr * ei + pi * er;
    pr = npr; pi = npi;
  }
  g[mat * 8192 + e] = pr;
  g[mat * 8192 + 4096 + e] = pi;
}

// ------------------- wave-level fragment helpers ---------------------------
// A 16x4 f32 layout: lanes 0-15 rows M=0..15, VGPR0/1 = K{0,1} (lanes<16) or
// K{2,3} (lanes>=16).  B 4x16: VGPR v holds row K=v (lanes 0-15) / K=v+2.
// C/D: VGPR n -> M=n (lanes 0-15) / M=n+8 (lanes 16-31), N = lane&15.

// ---------------------------------------------------------------------------
// Lo-stage, real input:  Y[(h,l), j] = sum_l' Ulo[l,l'] * X[j, h*64+l']
// grid(x=256 col-tiles of 16, y=64 h-blocks), 128 threads (4 waves).
// ---------------------------------------------------------------------------
__global__ __launch_bounds__(128) void pass_lo_real_kernel(
    const float* __restrict__ X, const float* __restrict__ gates,
    float* __restrict__ out) {
  __shared__ float sGr[64 * 65];
  __shared__ float sGi[64 * 65];
  __shared__ float sB[64 * 17];
  const int jt = blockIdx.x;
  const int h = blockIdx.y;
  const int tid = threadIdx.x;
  for (int i = tid; i < 4096; i += 128) {
    int rr = i >> 6, cc = i & 63;
    sGr[rr * 65 + cc] = gates[i];            // Glo_r
    sGi[rr * 65 + cc] = gates[4096 + i];     // Glo_i
  }
  // B[k,c] = X[(jt*16+c)*DIM + h*64 + k]  (k contiguous in memory -> coalesced)
  for (int i = tid; i < 1024; i += 128) {
    int c = i >> 6, k = i & 63;
    sB[k * 17 + c] = X[(size_t)(jt * 16 + c) * DIM + h * 64 + k];
  }
  __syncthreads();

  const int wave = tid >> 5;
  const int lane = tid & 31;
  const int l16 = lane & 15;
  const int half = lane >> 4;
  const int aRow = wave * 16 + l16;
  v8f accR = {};
  v8f accI = {};
#pragma unroll
  for (int kb = 0; kb < 16; ++kb) {
    const int kA = kb * 4 + half * 2;
    v2f ar = {sGr[aRow * 65 + kA], sGr[aRow * 65 + kA + 1]};
    v2f ai = {sGi[aRow * 65 + kA], sGi[aRow * 65 + kA + 1]};
    v2f b = {sB[kA * 17 + l16], sB[(kA + 1) * 17 + l16]};
    accR = wmma4(ar, b, accR);  // real part: Ur * X
    accI = wmma4(ai, b, accI);  // imag part: Ui * X
  }
#pragma unroll
  for (int n = 0; n < 8; ++n) {
    int l = wave * 16 + n + 8 * half;
    size_t off = ((size_t)(h * 64 + l) * DIM + jt * 16 + l16) * 2;
    out[off] = accR[n];
    out[off + 1] = accI[n];
  }
}

// ---------------------------------------------------------------------------
// Lo-stage, conj-transposed complex input:
//   Z[(h,l), j] = sum_l' Ulo[l,l'] * conj(W[j, h*64+l'])
// ---------------------------------------------------------------------------
__global__ __launch_bounds__(128) void pass_lo_conj_kernel(
    const float* __restrict__ Win, const float* __restrict__ gates,
    float* __restrict__ out) {
  __shared__ float sGr[64 * 65];
  __shared__ float sGi[64 * 65];
  __shared__ float sBr[64 * 17];
  __shared__ float sBi[64 * 17];
  const int jt = blockIdx.x;
  const int h = blockIdx.y;
  const int tid = threadIdx.x;
  for (int i = tid; i < 4096; i += 128) {
    int rr = i >> 6, cc = i & 63;
    sGr[rr * 65 + cc] = gates[i];
    sGi[rr * 65 + cc] = gates[4096 + i];
  }
  // B[k,c] = conj(W[(jt*16+c), h*64+k])  -- 64 contiguous complex per row
  for (int i = tid; i < 1024; i += 128) {
    int c = i >> 6, k = i & 63;
    size_t off = ((size_t)(jt * 16 + c) * DIM + h * 64 + k) * 2;
    sBr[k * 17 + c] = Win[off];
    sBi[k * 17 + c] = -Win[off + 1];
  }
  __syncthreads();

  const int wave = tid >> 5;
  const int lane = tid & 31;
  const int l16 = lane & 15;
  const int half = lane >> 4;
  const int aRow = wave * 16 + l16;
  v8f accR = {};
  v8f accI = {};
#pragma unroll
  for (int kb = 0; kb < 16; ++kb) {
    const int kA = kb * 4 + half * 2;
    v2f ar = {sGr[aRow * 65 + kA], sGr[aRow * 65 + kA + 1]};
    v2f ai = {sGi[aRow * 65 + kA], sGi[aRow * 65 + kA + 1]};
    v2f ain = -ai;
    v2f br = {sBr[kA * 17 + l16], sBr[(kA + 1) * 17 + l16]};
    v2f bi = {sBi[kA * 17 + l16], sBi[(kA + 1) * 17 + l16]};
    accR = wmma4(ar, br, accR);   // + Ar*Br
    accR = wmma4(ain, bi, accR);  // - Ai*Bi
    accI = wmma4(ar, bi, accI);   // + Ar*Bi
    accI = wmma4(ai, br, accI);   // + Ai*Br
  }
#pragma unroll
  for (int n = 0; n < 8; ++n) {
    int l = wave * 16 + n + 8 * half;
    size_t off = ((size_t)(h * 64 + l) * DIM + jt * 16 + l16) * 2;
    out[off] = accR[n];
    out[off + 1] = accI[n];
  }
}

// ---------------------------------------------------------------------------
// Hi-stage (complex):  Y[(h,l), j] = sum_h' Uhi[h,h'] * Yin[(h'*64+l), j]
// grid(x=256 col-tiles, y=64 l values), 128 threads.
// ---------------------------------------------------------------------------
__global__ __launch_bounds__(128) void pass_hi_kernel(
    const float* __restrict__ Yin, const float* __restrict__ gates,
    float* __restrict__ out) {
  __shared__ float sGr[64 * 65];
  __shared__ float sGi[64 * 65];
  __shared__ float sBr[64 * 17];
  __shared__ float sBi[64 * 17];
  const int jt = blockIdx.x;
  const int l = blockIdx.y;
  const int tid = threadIdx.x;
  for (int i = tid; i < 4096; i += 128) {
    int rr = i >> 6, cc = i & 63;
    sGr[rr * 65 + cc] = gates[8192 + i];      // Ghi_r
    sGi[rr * 65 + cc] = gates[12288 + i];     // Ghi_i
  }
  // B[k,c] = Yin[(k*64+l)*DIM + jt*16+c]  (16 complex = 128B contiguous/row)
  for (int i = tid; i < 1024; i += 128) {
    int k = i >> 4, c = i & 15;
    size_t off = ((size_t)(k * 64 + l) * DIM + jt * 16 + c) * 2;
    sBr[k * 17 + c] = Yin[off];
    sBi[k * 17 + c] = Yin[off + 1];
  }
  __syncthreads();

  const int wave = tid >> 5;
  const int lane = tid & 31;
  const int l16 = lane & 15;
  const int half = lane >> 4;
  const int aRow = wave * 16 + l16;
  v8f accR = {};
  v8f accI = {};
#pragma unroll
  for (int kb = 0; kb < 16; ++kb) {
    const int kA = kb * 4 + half * 2;
    v2f ar = {sGr[aRow * 65 + kA], sGr[aRow * 65 + kA + 1]};
    v2f ai = {sGi[aRow * 65 + kA], sGi[aRow * 65 + kA + 1]};
    v2f ain = -ai;
    v2f br = {sBr[kA * 17 + l16], sBr[(kA + 1) * 17 + l16]};
    v2f bi = {sBi[kA * 17 + l16], sBi[(kA + 1) * 17 + l16]};
    accR = wmma4(ar, br, accR);
    accR = wmma4(ain, bi, accR);
    accI = wmma4(ar, bi, accI);
    accI = wmma4(ai, br, accI);
  }
#pragma unroll
  for (int n = 0; n < 8; ++n) {
    int hh = wave * 16 + n + 8 * half;
    size_t off = ((size_t)(hh * 64 + l) * DIM + jt * 16 + l16) * 2;
    out[off] = accR[n];
    out[off + 1] = accI[n];
  }
}

// ---------------------------------------------------------------------------
extern "C" void kernel_launch(void* const* d_in, const int* in_sizes, int n_in,
                              void* d_out, int out_size, void* d_ws,
                              size_t ws_size, hipStream_t stream) {
  const float* weight = (const float*)d_in[0]; // 36 angles
  const float* x = (const float*)d_in[1];      // 4096x4096 f32
  float* out = (float*)d_out;                  // 4096x4096 complex64 (interleaved)

  // ws: [0, 16384) floats -> gate factors, then one full complex ping buffer.
  float* gates = (float*)d_ws;
  float* ybuf = gates + 16384; // needs DIM*DIM*2 floats (134 MB)

  build_gates_kernel<<<32, 256, 0, stream>>>(weight, gates);

  dim3 grid(DIM / 16, 64);
  dim3 blk(128);
  // Pass A: W = U * X^T        (lo then hi); W lands in d_out (temp).
  pass_lo_real_kernel<<<grid, blk, 0, stream>>>(x, gates, ybuf);
  pass_hi_kernel<<<grid, blk, 0, stream>>>(ybuf, gates, out);
  // Pass B: Out = U * W^H      (lo-conj then hi); final result in d_out.
  pass_lo_conj_kernel<<<grid, blk, 0, stream>>>(out, gates, ybuf);
  pass_hi_kernel<<<grid, blk, 0, stream>>>(ybuf, gates, out);
}